// NeighborSample_7464653161010
// MI455X (gfx1250) — compile-verified
//
#include <hip/hip_runtime.h>

// NeighborSample: x[8,64,64,192] f32 -> out[8*64*64, 5, 5, 192] f32
// Pure data movement; store-bandwidth-bound (~629 MB out @ 23.3 TB/s -> ~27us floor).
// CDNA5 path: stage 5 input rows per (b,y) into LDS with the Tensor Data Mover
// (tensor_load_to_lds, TENSORcnt), then emit the 25 shifted copies from LDS
// with fully coalesced 128-bit stores (halo -> zeros).

#define KNB 5
#define H 64
#define W 64
#define C 192
#define C4 (C / 4)                    // 48 float4 per pixel
#define ROW_F4 (W * C4)               // 3072 float4 per input row
#define ROW_ELTS (W * C)              // 12288 f32 per input row
#define ROW_BYTES (ROW_ELTS * 4)      // 49152 B per input row
#define PIX_OUT_F4 (KNB * KNB * C4)   // 1200 float4 per output pixel
#define WG_OUT_F4 (W * PIX_OUT_F4)    // 76800 float4 per workgroup
#define LDS_BYTES (KNB * ROW_BYTES)   // 245760 B (fits 320 KB WGP LDS)
#define NTHREADS 512

typedef unsigned int u32;
typedef unsigned long long u64;
typedef __attribute__((ext_vector_type(4))) u32 u32x4;
typedef __attribute__((ext_vector_type(8))) u32 u32x8;

__global__ void __launch_bounds__(NTHREADS)
neighbor_sample_kernel(const float4* __restrict__ x4, float4* __restrict__ out4) {
  extern __shared__ float4 lds4[];

  const int wg  = blockIdx.x;     // 0..511
  const int b   = wg >> 6;        // batch
  const int y   = wg & 63;        // row
  const int tid = threadIdx.x;

  // ---- Stage rows y-2..y+2 of x[b] into LDS via the Tensor Data Mover ----
  // Waves 0..4 each DMA one 48 KB row. readfirstlane makes the wave id a
  // scalar so the guard is a scalar branch (TDM ignores EXEC, so we must
  // branch around it, not mask it).
  const float4* batch_base = x4 + (size_t)b * (size_t)(H * ROW_F4);
  const int wave = __builtin_amdgcn_readfirstlane((int)threadIdx.x) >> 5;
  if (wave < KNB) {
    const int ysrc = y + wave - 2;
    if ((unsigned)ysrc < (unsigned)H) {   // scalar; halo rows simply not loaded
      const u64 gaddr = (u64)(const void*)(batch_base + (size_t)ysrc * ROW_F4);
      // Low 32 bits of the generic LDS pointer = LDS byte address.
      const u32 lds_byte = (u32)(size_t)(const void*)(&lds4[wave * ROW_F4]);

      // ---- D# group 0 (128b): count=1, lds_addr, global_addr[56:0], type=2
      u32x4 g0;
      g0.x = 1u;                                        // count=1, user mode
      g0.y = lds_byte;                                  // lds_addr
      g0.z = (u32)gaddr;                                // global_addr[31:0]
      g0.w = (u32)((gaddr >> 32) & 0x01FFFFFFull)       // global_addr[56:32]
           | (2u << 30);                                // type = 2 ("image")

      // ---- D# group 1 (256b): 4B elements, 1 row of ROW_ELTS, contiguous
      u32x8 g1;
      g1.s0 = 2u << 16;                                 // data_size=4B, mask=0
      g1.s1 = ((u32)ROW_ELTS & 0xFFFFu) << 16;          // tensor_dim0[15:0]
      g1.s2 = ((u32)ROW_ELTS >> 16)                     // tensor_dim0[31:16]
            | (1u << 16);                               // tensor_dim1 = 1
      g1.s3 = ((u32)ROW_ELTS & 0xFFFFu) << 16;          // tile_dim0 = ROW_ELTS
      g1.s4 = 1u;                                       // tile_dim1=1, tile_dim2=0
      g1.s5 = (u32)ROW_ELTS;                            // tensor_dim0_stride
      g1.s6 = 0u;                                       // stride0[47:32], stride1[15:0]
      g1.s7 = 0u;                                       // stride1[47:16]

      asm volatile("tensor_load_to_lds %0, %1"
                   :: "s"(g0), "s"(g1)
                   : "memory");
      __builtin_amdgcn_s_wait_tensorcnt(0);             // s_wait_tensorcnt 0
    }
  }
  __syncthreads();   // LDS rows visible to all 16 waves

  // ---- Emit 25 shifted copies: linear, fully coalesced b128 stores ----
  const size_t obase = (size_t)wg * (size_t)WG_OUT_F4;
  for (int o = tid; o < WG_OUT_F4; o += NTHREADS) {
    const int px = o / PIX_OUT_F4;             // pixel x within row
    const int r  = o - px * PIX_OUT_F4;        // 0..1199
    const int i  = r / (KNB * C4);             // neighborhood row 0..4
    const int rr = r - i * (KNB * C4);
    const int j  = rr / C4;                    // neighborhood col 0..4
    const int k4 = rr - j * C4;                // channel float4 0..47

    const int col = px + j - 2;
    const int row = y + i - 2;

    float4 v = make_float4(0.f, 0.f, 0.f, 0.f);
    if (((unsigned)col < (unsigned)W) & ((unsigned)row < (unsigned)H)) {
      v = lds4[i * ROW_F4 + col * C4 + k4];    // ds_load_b128
    }
    out4[obase + o] = v;                        // global_store_b128
  }
}

extern "C" void kernel_launch(void* const* d_in, const int* in_sizes, int n_in,
                              void* d_out, int out_size, void* d_ws, size_t ws_size,
                              hipStream_t stream) {
  (void)in_sizes; (void)n_in; (void)d_ws; (void)ws_size; (void)out_size;
  const float4* x4 = (const float4*)d_in[0];
  float4* out4 = (float4*)d_out;

  // Allow 240 KB of dynamic LDS (beyond the 64 KB default cap).
  hipFuncSetAttribute(reinterpret_cast<const void*>(neighbor_sample_kernel),
                      hipFuncAttributeMaxDynamicSharedMemorySize, LDS_BYTES);

  dim3 grid(8 * H);       // one workgroup per (batch, row) = 512
  dim3 block(NTHREADS);
  neighbor_sample_kernel<<<grid, block, LDS_BYTES, stream>>>(x4, out4);
}